// ConceptHead_24318104830230
// MI455X (gfx1250) — compile-verified
//
#include <hip/hip_runtime.h>
#include <hip/hip_bf16.h>

// ---------------------------------------------------------------------------
// ConceptHead: logits GEMM (f32 WMMA) + streaming top-16 + weighted embedding
// gather, for MI455X / gfx1250 (wave32, V_WMMA_F32_16X16X4_F32).
// ---------------------------------------------------------------------------

typedef __attribute__((ext_vector_type(2))) float v2f;
typedef __attribute__((ext_vector_type(8))) float v8f;

static constexpr int kB    = 2;
static constexpr int kT    = 1024;
static constexpr int kBT   = kB * kT;      // 2048 token rows
static constexpr int kD    = 1024;         // feature dim (GEMM K)
static constexpr int kC    = 16384;        // concepts (GEMM N)
static constexpr int kKGT  = 8;            // ground-truth concepts per token
static constexpr int kTOPK = 16;

static constexpr int kKC      = 64;            // K chunk staged in LDS
static constexpr int kColsWG  = 512;           // columns per workgroup
static constexpr int kColsIt  = 128;           // columns per inner iteration (8 waves * 16)
static constexpr int kNIt     = kColsWG / kColsIt;  // 4
static constexpr int kNBlk    = kC / kColsWG;       // 32 column blocks

// ---------------------------------------------------------------------------
// Stage 1: per (16-row tile, 512-col block) compute logits with f32 WMMA and
// keep per-row top-16 candidates. Writes cand_l / cand_i to workspace.
// ---------------------------------------------------------------------------
__global__ __launch_bounds__(256) void concept_logits_topk_kernel(
    const float* __restrict__ hidden,   // (BT, D)
    const float* __restrict__ wgt,      // (C, D)
    float* __restrict__ cand_l,         // (BT, NBLK, TOPK)
    int*   __restrict__ cand_i)         // (BT, NBLK, TOPK)
{
    __shared__ float As[16][kKC + 2];        // 16x64 A tile (+pad)
    __shared__ float Bs[kColsIt][kKC + 2];   // 128x64 B tile (+pad)
    __shared__ float Ls[kColsIt][16];        // logits, transposed [col][row]

    const int tid   = threadIdx.x;
    const int lane  = tid & 31;
    const int wave  = tid >> 5;              // 0..7
    const int nloc  = lane & 15;             // N (and A's M) within 16
    const int khalf = lane >> 4;             // 0 or 1 (K half for 16x16x4)

    const int row0    = blockIdx.x * 16;     // token-row tile
    const int cb      = blockIdx.y;          // column block
    const int col_wg0 = cb * kColsWG;

    // per-row top-16 state, sorted descending; lives on lanes 0..15 of wave 0
    float bl[kTOPK];
    int   bi[kTOPK];
#pragma unroll
    for (int j = 0; j < kTOPK; ++j) { bl[j] = -3.4e38f; bi[j] = 0; }

    for (int it = 0; it < kNIt; ++it) {
        const int colbase = col_wg0 + it * kColsIt;
        v8f acc = {};

        for (int kc = 0; kc < kD; kc += kKC) {
            // ---- stage A tile: 16 rows x 64 floats, 4 floats per thread
            {
                const int r = tid >> 4;             // 0..15
                const int c = (tid & 15) << 2;      // 0..60
                const float4 av =
                    *(const float4*)(hidden + (size_t)(row0 + r) * kD + kc + c);
                As[r][c + 0] = av.x; As[r][c + 1] = av.y;
                As[r][c + 2] = av.z; As[r][c + 3] = av.w;
            }
            // ---- stage B tile: 128 rows x 64 floats (coalesced b128 loads)
#pragma unroll
            for (int p = 0; p < 8; ++p) {
                const int r = (tid >> 4) + p * 16;  // 0..127
                const int c = (tid & 15) << 2;
                const float4 bv =
                    *(const float4*)(wgt + (size_t)(colbase + r) * kD + kc + c);
                Bs[r][c + 0] = bv.x; Bs[r][c + 1] = bv.y;
                Bs[r][c + 2] = bv.z; Bs[r][c + 3] = bv.w;
            }
            __syncthreads();

            // ---- 16 WMMA steps over this K chunk
            const int colloc = (wave << 4) + nloc;
#pragma unroll
            for (int k = 0; k < kKC; k += 4) {
                const v2f a = *(const v2f*)(&As[nloc][k + 2 * khalf]);
                const v2f b = *(const v2f*)(&Bs[colloc][k + 2 * khalf]);
                acc = __builtin_amdgcn_wmma_f32_16x16x4_f32(
                    false, a, false, b, (short)0, acc, false, false);
            }
            __syncthreads();
        }

        // ---- dump 16x16 tile transposed into LDS: VGPR j -> row j + 8*khalf
        {
            const int colloc = (wave << 4) + nloc;
#pragma unroll
            for (int j = 0; j < 8; ++j)
                Ls[colloc][j + 8 * khalf] = acc[j];
        }
        __syncthreads();

        // ---- per-row top-16 update (lanes 0..15 of wave 0)
        if (tid < 16) {
            for (int c2 = 0; c2 < kColsIt; ++c2) {
                const float v = Ls[c2][tid];
                if (v > bl[kTOPK - 1]) {
                    float cl = v;
                    int   ci = colbase + c2;
#pragma unroll
                    for (int j = 0; j < kTOPK; ++j) {
                        if (cl > bl[j]) {
                            float tl = bl[j]; bl[j] = cl; cl = tl;
                            int   ti = bi[j]; bi[j] = ci; ci = ti;
                        }
                    }
                }
            }
        }
        __syncthreads();
    }

    if (tid < 16) {
        const size_t base = ((size_t)(row0 + tid) * kNBlk + cb) * kTOPK;
#pragma unroll
        for (int j = 0; j < kTOPK; ++j) {
            cand_l[base + j] = bl[j];
            cand_i[base + j] = bi[j];
        }
    }
}

// ---------------------------------------------------------------------------
// Stage 2: merge 32*16 candidates -> global top-16, sigmoid weights, add
// ground-truth pooled embeddings, weighted gather over concept_emb.
// ---------------------------------------------------------------------------
__global__ __launch_bounds__(256) void concept_combine_kernel(
    const float* __restrict__ emb,       // (C, D)
    const int*   __restrict__ ids,       // (BT, K_GT)
    const unsigned char* __restrict__ mask,  // (BT, K_GT) bool
    const float* __restrict__ cand_l,
    const int*   __restrict__ cand_i,
    float* __restrict__ out)             // (BT, D)
{
    const int row = blockIdx.x;
    __shared__ float s_c[kTOPK + kKGT];
    __shared__ int   s_id[kTOPK + kKGT];

    if (threadIdx.x == 0) {
        float bl[kTOPK];
        int   bi[kTOPK];
#pragma unroll
        for (int j = 0; j < kTOPK; ++j) { bl[j] = -3.4e38f; bi[j] = 0; }

        const float* cl = cand_l + (size_t)row * (kNBlk * kTOPK);
        const int*   ci = cand_i + (size_t)row * (kNBlk * kTOPK);
        for (int e = 0; e < kNBlk * kTOPK; ++e) {
            const float v = cl[e];
            if (v > bl[kTOPK - 1]) {
                float c = v;
                int   ix = ci[e];
#pragma unroll
                for (int j = 0; j < kTOPK; ++j) {
                    if (c > bl[j]) {
                        float tl = bl[j]; bl[j] = c; c = tl;
                        int   ti = bi[j]; bi[j] = ix; ix = ti;
                    }
                }
            }
        }
#pragma unroll
        for (int j = 0; j < kTOPK; ++j) {
            // 0.5 * sigmoid(logit)   (TF_ALPHA = 0.5)
            s_c[j]  = 0.5f / (1.0f + __expf(-bl[j]));
            s_id[j] = bi[j];
        }
#pragma unroll
        for (int j = 0; j < kKGT; ++j) {
            const int id  = ids[(size_t)row * kKGT + j];
            const bool va = (mask[(size_t)row * kKGT + j] != 0) && (id != -1);
            s_c[kTOPK + j]  = va ? 0.5f : 0.0f;
            s_id[kTOPK + j] = va ? id : 0;
        }
    }
    __syncthreads();

    const int d = threadIdx.x << 2;   // 256 threads * 4 floats = 1024 = D
    float ax = 0.f, ay = 0.f, az = 0.f, aw = 0.f;
#pragma unroll
    for (int e = 0; e < kTOPK + kKGT; ++e) {
        const float c = s_c[e];
        const float4 ev = *(const float4*)(emb + (size_t)s_id[e] * kD + d);
        ax += c * ev.x; ay += c * ev.y; az += c * ev.z; aw += c * ev.w;
    }
    float4 o; o.x = ax; o.y = ay; o.z = az; o.w = aw;
    *(float4*)(out + (size_t)row * kD + d) = o;
}

// ---------------------------------------------------------------------------
extern "C" void kernel_launch(void* const* d_in, const int* in_sizes, int n_in,
                              void* d_out, int out_size, void* d_ws, size_t ws_size,
                              hipStream_t stream) {
    const float* hidden = (const float*)d_in[0];   // (B,T,D) f32
    const float* wgt    = (const float*)d_in[1];   // (C,D)   f32
    const float* emb    = (const float*)d_in[2];   // (C,D)   f32
    const int*   ids    = (const int*)d_in[3];     // (B,T,K_GT) i32
    const unsigned char* mask = (const unsigned char*)d_in[4];  // bool
    float* out = (float*)d_out;

    float* cand_l = (float*)d_ws;
    int*   cand_i = (int*)((char*)d_ws +
                           (size_t)kBT * kNBlk * kTOPK * sizeof(float));

    dim3 g1(kBT / 16, kNBlk);
    concept_logits_topk_kernel<<<g1, 256, 0, stream>>>(hidden, wgt, cand_l, cand_i);
    concept_combine_kernel<<<kBT, 256, 0, stream>>>(emb, ids, mask,
                                                    cand_l, cand_i, out);
}